// FairGNN_51917564674443
// MI455X (gfx1250) — compile-verified
//
#include <hip/hip_runtime.h>

typedef __attribute__((ext_vector_type(2))) float v2f;
typedef __attribute__((ext_vector_type(8))) float v8f;
typedef __attribute__((ext_vector_type(4))) int   v4i;

#define NFEAT 256
#define NHID  64
#define LDSW  260   // padded row stride (floats) to avoid 64-bank conflicts

#if __has_builtin(__builtin_amdgcn_global_load_async_to_lds_b128)
#define HAVE_ASYNC_LDS 1
#else
#define HAVE_ASYNC_LDS 0
#endif

typedef __attribute__((address_space(1))) v4i* gptr_v4i;
typedef __attribute__((address_space(3))) v4i* lptr_v4i;

__device__ __forceinline__ void wait_async_zero() {
#if __has_builtin(__builtin_amdgcn_s_wait_asynccnt)
    __builtin_amdgcn_s_wait_asynccnt(0);
#else
    asm volatile("s_wait_asynccnt 0x0" ::: "memory");
#endif
}

// ---------------- degree / normalization ----------------
__global__ void deg_init_kernel(float* __restrict__ deg, int n) {
    int i = blockIdx.x * blockDim.x + threadIdx.x;
    if (i < n) deg[i] = 1.0f;             // self-loop contributes 1 to every node
}

__global__ void deg_accum_kernel(const int* __restrict__ col, float* __restrict__ deg, int e) {
    int i = blockIdx.x * blockDim.x + threadIdx.x;
    if (i < e) atomicAdd(&deg[col[i]], 1.0f);
}

__global__ void deg_rsqrt_kernel(float* __restrict__ deg, int n) {
    int i = blockIdx.x * blockDim.x + threadIdx.x;
    if (i < n) deg[i] = rsqrtf(deg[i]);   // deg >= 1 always, matches reference guard
}

// ---------------- fused WMMA GEMM: h = x @ W_gnn, hs = x @ W_est ----------------
// 128 threads = 4 waves; each wave -> one 16x16 tile of the 16x64 output strip.
__global__ __launch_bounds__(128) void gemm_wmma_kernel(
    const float* __restrict__ x, const float* __restrict__ Wg,
    const float* __restrict__ We, float* __restrict__ h,
    float* __restrict__ hs, int nNodes)
{
    __shared__ float xs[16 * LDSW];
    const int tid  = threadIdx.x;
    const int m0   = blockIdx.x * 16;
    const bool full = (m0 + 16 <= nNodes);

    // Stage 16x256 x-tile into LDS (x is read from HBM exactly once).
#if HAVE_ASYNC_LDS
    if (full) {
        // Direct global->LDS DMA, 16B per lane-transfer, tracked by ASYNCcnt.
        #pragma unroll
        for (int it = 0; it < 8; ++it) {
            int idx = tid + it * 128;          // 0..1023 b128 transfers
            int r   = idx >> 6;                // row 0..15
            int c0  = (idx & 63) << 2;         // col 0,4,...,252
            __builtin_amdgcn_global_load_async_to_lds_b128(
                (gptr_v4i)(x + (size_t)(m0 + r) * NFEAT + c0),
                (lptr_v4i)&xs[r * LDSW + c0],
                0, 0);
        }
        wait_async_zero();
    } else
#endif
    {
        for (int idx = tid; idx < 16 * NFEAT; idx += 128) {
            int r = idx >> 8, c = idx & (NFEAT - 1);
            int m = m0 + r;
            xs[r * LDSW + c] = (m < nNodes) ? x[(size_t)m * NFEAT + c] : 0.0f;
        }
    }
    __syncthreads();

    const int wave = tid >> 5;
    const int lane = tid & 31;
    const int half = lane >> 4;     // K-half selector for 32-bit A/B fragments
    const int l16  = lane & 15;
    const int n0   = wave * 16;     // this wave's 16-column slice of W_gnn

    v8f acc = {};
    for (int kb = 0; kb < NFEAT; kb += 4) {
        const int k = kb + 2 * half;
        v2f a, b;
        // A 16x4 f32 frag: lane l16 = row, VGPR j = K within half (ISA 7.12.2)
        a.x = xs[l16 * LDSW + k];
        a.y = xs[l16 * LDSW + k + 1];
        // B 4x16 f32 frag: lane l16 = col, same K striping
        b.x = Wg[(size_t)k * NHID + n0 + l16];
        b.y = Wg[(size_t)(k + 1) * NHID + n0 + l16];
        acc = __builtin_amdgcn_wmma_f32_16x16x4_f32(
            /*neg_a=*/false, a, /*neg_b=*/false, b,
            /*c_mod=*/(short)0, acc, /*reuse_a=*/false, /*reuse_b=*/false);
    }

    // D 16x16 f32: VGPR v -> row v + 8*half, col l16. Uniform tile guard only.
    if (full) {
        #pragma unroll
        for (int v = 0; v < 8; ++v)
            h[(size_t)(m0 + v + 8 * half) * NHID + n0 + l16] = acc[v];
    } else {
        for (int v = 0; v < 8; ++v) {
            int m = m0 + v + 8 * half;
            if (m < nNodes) h[(size_t)m * NHID + n0 + l16] = acc[v];
        }
    }

    // Fused estimator matvec from the already-staged LDS tile
    if (tid < 16 && (m0 + tid) < nNodes) {
        float e = 0.0f;
        for (int k = 0; k < NFEAT; ++k) e += xs[tid * LDSW + k] * We[k];
        hs[m0 + tid] = e;
    }
}

// ---------------- init: bias + self-loop term (deterministic base) ----------------
__global__ void init_out_kernel(const float* __restrict__ dinv,
                                const float* __restrict__ h,
                                const float* __restrict__ hs,
                                const float* __restrict__ b_gnn,
                                const float* __restrict__ b_est,
                                float* __restrict__ z, float* __restrict__ s,
                                int nNodes)
{
    int tid = blockIdx.x * blockDim.x + threadIdx.x;
    if (tid >= nNodes * NHID) return;
    int i = tid >> 6, f = tid & 63;
    float d = dinv[i];
    float d2 = d * d;                       // self-loop norm = dinv[i]*dinv[i]
    z[tid] = b_gnn[f] + d2 * h[tid];
    if (f == 0) s[i] = b_est[0] + d2 * hs[i];
}

// ---------------- edge scatter (the L2-atomic-bound hot loop) ----------------
// One wave32 per edge: gather 64-float h[row], 2 atomic f32 adds per lane into z[col],
// lane 0 handles the scalar estimator channel. h + z (~51 MB) stay resident in L2.
__global__ __launch_bounds__(256) void edge_scatter_kernel(
    const int* __restrict__ row, const int* __restrict__ col,
    const float* __restrict__ dinv, const float* __restrict__ h,
    const float* __restrict__ hs, float* __restrict__ z,
    float* __restrict__ s, int nEdges)
{
    int edge = (blockIdx.x * 256 + threadIdx.x) >> 5;
    int lane = threadIdx.x & 31;
    if (edge >= nEdges) return;
    int r = row[edge];
    int c = col[edge];
    float nrm = dinv[r] * dinv[c];
    const float* hrow = h + (size_t)r * NHID;
    float v0 = hrow[lane];
    float v1 = hrow[lane + 32];
    float* zrow = z + (size_t)c * NHID;
    atomicAdd(&zrow[lane],      nrm * v0);
    atomicAdd(&zrow[lane + 32], nrm * v1);
    if (lane == 0) atomicAdd(&s[c], nrm * hs[r]);
}

// ---------------- classifier head: y = z @ W_cls + b_cls ----------------
__global__ __launch_bounds__(256) void cls_kernel(const float* __restrict__ z,
                                                  const float* __restrict__ Wc,
                                                  const float* __restrict__ bc,
                                                  float* __restrict__ y, int nNodes)
{
    int node = (blockIdx.x * 256 + threadIdx.x) >> 5;
    int lane = threadIdx.x & 31;
    if (node >= nNodes) return;
    const float* zr = z + (size_t)node * NHID;
    float acc = zr[lane] * Wc[lane] + zr[lane + 32] * Wc[lane + 32];
    #pragma unroll
    for (int off = 16; off > 0; off >>= 1)
        acc += __shfl_xor(acc, off, 32);
    if (lane == 0) y[node] = acc + bc[0];
}

extern "C" void kernel_launch(void* const* d_in, const int* in_sizes, int n_in,
                              void* d_out, int out_size, void* d_ws, size_t ws_size,
                              hipStream_t stream) {
    const int*   edge  = (const int*)d_in[0];   // [2, E] flat: sources then targets
    const float* x     = (const float*)d_in[1]; // [N, 256]
    const float* W_est = (const float*)d_in[2]; // [256, 1]
    const float* b_est = (const float*)d_in[3]; // [1]
    const float* W_gnn = (const float*)d_in[4]; // [256, 64]
    const float* b_gnn = (const float*)d_in[5]; // [64]
    const float* W_cls = (const float*)d_in[6]; // [64, 1]
    const float* b_cls = (const float*)d_in[7]; // [1]

    const int E = in_sizes[0] / 2;
    const int N = in_sizes[1] / NFEAT;

    const int* rowIdx = edge;
    const int* colIdx = edge + E;

    // workspace layout (floats): dinv[N] | hs[N] | h[N*64] | z[N*64]  (~52 MB)
    float* wsf  = (float*)d_ws;
    float* dinv = wsf;
    float* hs   = wsf + (size_t)N;
    float* h    = wsf + 2 * (size_t)N;
    float* z    = h + (size_t)N * NHID;

    float* y = (float*)d_out;        // output 0: y [N]
    float* s = (float*)d_out + N;    // output 1: s [N]

    deg_init_kernel <<<(N + 255) / 256, 256, 0, stream>>>(dinv, N);
    deg_accum_kernel<<<(E + 255) / 256, 256, 0, stream>>>(colIdx, dinv, E);
    deg_rsqrt_kernel<<<(N + 255) / 256, 256, 0, stream>>>(dinv, N);

    gemm_wmma_kernel<<<(N + 15) / 16, 128, 0, stream>>>(x, W_gnn, W_est, h, hs, N);

    init_out_kernel<<<((size_t)N * NHID + 255) / 256, 256, 0, stream>>>(
        dinv, h, hs, b_gnn, b_est, z, s, N);

    edge_scatter_kernel<<<(E + 7) / 8, 256, 0, stream>>>(
        rowIdx, colIdx, dinv, h, hs, z, s, E);

    cls_kernel<<<(N + 7) / 8, 256, 0, stream>>>(z, W_cls, b_cls, y, N);
}